// Voxelization_76922864272024
// MI455X (gfx1250) — compile-verified
//
#include <hip/hip_runtime.h>
#include <hip/hip_bf16.h>

// Problem constants (from reference setup_inputs / R=32)
#define BB 16
#define CC 64
#define NN 65536
#define RR 32
#define R3 32768                      // 32^3
#define VOX_TOTAL (BB * CC * R3)      // 33,554,432 floats
#define COORD_TOTAL (BB * 3 * NN)     // 3,145,728 floats

typedef __attribute__((ext_vector_type(2))) float v2f;
typedef __attribute__((ext_vector_type(8))) float v8f;

// ---------------------------------------------------------------------------
// Kernel 0: zero the vox accumulator region of d_out and the int counts in ws.
// ---------------------------------------------------------------------------
__global__ __launch_bounds__(256) void vx_zero_kernel(float4* __restrict__ vox4,
                                                      int4* __restrict__ cnt4) {
    const int gid = blockIdx.x * 256 + threadIdx.x;
    // VOX_TOTAL/4 = 8,388,608 float4s
    if (gid < (VOX_TOTAL / 4)) {
        float4 z; z.x = 0.f; z.y = 0.f; z.z = 0.f; z.w = 0.f;
        vox4[gid] = z;
    }
    // BB*R3/4 = 131,072 int4s
    if (gid < (BB * R3 / 4)) {
        int4 zi; zi.x = 0; zi.y = 0; zi.z = 0; zi.w = 0;
        cnt4[gid] = zi;
    }
}

// ---------------------------------------------------------------------------
// Kernel 1: per-batch per-axis mean of points, reduced on the matrix pipe.
// A = ones(16x4) fp32  ->  D[m][n] += sum_k B[k][n]  (exact fp32 column sums).
// We sum ALL 16 columns at the end, so the result is invariant to the exact
// B-operand VGPR layout. Fixed trip count (1024 chunks / 8 waves = 128) gives
// a scalar-controlled uniform loop: EXEC is statically all-ones at every WMMA.
// ---------------------------------------------------------------------------
__global__ __launch_bounds__(256) void vx_mean_kernel(const float* __restrict__ pts,
                                                      float* __restrict__ means) {
    const int b    = blockIdx.x;
    const int tid  = threadIdx.x;
    const int wave = tid >> 5;
    const int lane = tid & 31;

    const float* p = pts + (size_t)b * NN * 3;

    v2f onesA; onesA[0] = 1.0f; onesA[1] = 1.0f;
    v8f accx = {}; v8f accy = {}; v8f accz = {};

    // Intended B[k][n] slots: VGPR0 = {K0 | K2}, VGPR1 = {K1 | K3} halves.
    const int e0 = lane + (lane & 16);   // lanes 0-15 -> 0..15 ; lanes 16-31 -> 32..47
    const int e1 = e0 + 16;              //                16..31 ;               48..63

    // Per-lane base byte element index for this wave's first chunk.
    const float* p0 = p + (size_t)(wave * 64 + e0) * 3;   // element i0 of chunk
    const float* p1 = p + (size_t)(wave * 64 + e1) * 3;   // element i1 of chunk

    const int NITER = (NN / 64) / 8;     // 128, compile-time constant
    #pragma unroll 2
    for (int it = 0; it < NITER; ++it) {
        const size_t step = (size_t)it * (8 * 64) * 3;    // 8 waves * 64 pts * 3 floats
        v2f bx; bx[0] = p0[step + 0]; bx[1] = p1[step + 0];
        v2f by; by[0] = p0[step + 1]; by[1] = p1[step + 1];
        v2f bz; bz[0] = p0[step + 2]; bz[1] = p1[step + 2];
        accx = __builtin_amdgcn_wmma_f32_16x16x4_f32(false, onesA, false, bx,
                                                     (short)0, accx, false, false);
        accy = __builtin_amdgcn_wmma_f32_16x16x4_f32(false, onesA, false, by,
                                                     (short)0, accy, false, false);
        accz = __builtin_amdgcn_wmma_f32_16x16x4_f32(false, onesA, false, bz,
                                                     (short)0, accz, false, false);
    }

    // D VGPR0, lanes 0-15 = row M=0 -> the 16 column sums.
    __shared__ float red[8 * 48];
    if (lane < 16) {
        red[wave * 48 + 0 * 16 + lane] = accx[0];
        red[wave * 48 + 1 * 16 + lane] = accy[0];
        red[wave * 48 + 2 * 16 + lane] = accz[0];
    }
    __syncthreads();
    if (tid < 3) {
        float s = 0.0f;
        for (int w = 0; w < 8; ++w)
            for (int n = 0; n < 16; ++n)
                s += red[w * 48 + tid * 16 + n];
        means[b * 3 + tid] = s / (float)NN;
    }
}

// ---------------------------------------------------------------------------
// Kernel 2: coords output + flat voxel index + per-voxel point counts.
// coord = clip(((p - mean) + 1) * 0.5 * R, 0, R-1); idx = round-half-even.
// ---------------------------------------------------------------------------
__global__ __launch_bounds__(256) void vx_coords_kernel(const float* __restrict__ pts,
                                                        const float* __restrict__ means,
                                                        float* __restrict__ coordsOut,
                                                        int* __restrict__ wsIdx,
                                                        int* __restrict__ wsCnt) {
    const int gid = blockIdx.x * 256 + threadIdx.x;   // over BB*NN
    const int b = gid >> 16;
    const int n = gid & (NN - 1);

    const float* p = pts + (size_t)gid * 3;
    const float mx = means[b * 3 + 0];
    const float my = means[b * 3 + 1];
    const float mz = means[b * 3 + 2];

    // ((p - m) + 1)/2 * 32 == (p - m + 1) * 16
    float cx = fminf(fmaxf((p[0] - mx + 1.0f) * 16.0f, 0.0f), 31.0f);
    float cy = fminf(fmaxf((p[1] - my + 1.0f) * 16.0f, 0.0f), 31.0f);
    float cz = fminf(fmaxf((p[2] - mz + 1.0f) * 16.0f, 0.0f), 31.0f);

    coordsOut[((size_t)b * 3 + 0) * NN + n] = cx;
    coordsOut[((size_t)b * 3 + 1) * NN + n] = cy;
    coordsOut[((size_t)b * 3 + 2) * NN + n] = cz;

    const int ix = (int)rintf(cx);   // v_rndne_f32: matches jnp.round (half-even)
    const int iy = (int)rintf(cy);
    const int iz = (int)rintf(cz);
    const int flat = (ix * RR + iy) * RR + iz;

    wsIdx[gid] = flat;
    atomicAdd(&wsCnt[b * R3 + flat], 1);
}

// ---------------------------------------------------------------------------
// Kernel 3: scatter-add features into voxels. Coalesced feature reads per c
// (consecutive n across lanes); fp32 atomics land in the L2-resident 128 MB
// vox region. global_prefetch_b8 hides the strided row hops.
// ---------------------------------------------------------------------------
__global__ __launch_bounds__(256) void vx_scatter_kernel(const float* __restrict__ feats,
                                                         const int* __restrict__ wsIdx,
                                                         float* __restrict__ vox) {
    const int gid = blockIdx.x * 256 + threadIdx.x;   // over BB*NN
    const int b = gid >> 16;
    const int n = gid & (NN - 1);

    const int flat = wsIdx[gid];
    const float* f = feats + (size_t)b * CC * NN + n;
    float* v = vox + (size_t)b * CC * R3 + flat;

    #pragma unroll 4
    for (int c = 0; c < CC; ++c) {
        if (c + 8 < CC)
            __builtin_prefetch(f + (size_t)(c + 8) * NN, 0, 0);
        unsafeAtomicAdd(v + (size_t)c * R3, f[(size_t)c * NN]);
    }
}

// ---------------------------------------------------------------------------
// Kernel 4: divide each voxel's feature vector by max(count, 1).
// ---------------------------------------------------------------------------
__global__ __launch_bounds__(256) void vx_norm_kernel(const int* __restrict__ wsCnt,
                                                      float* __restrict__ vox) {
    const int gid = blockIdx.x * 256 + threadIdx.x;   // over BB*R3
    const int b = gid >> 15;
    const int vv = gid & (R3 - 1);

    const int cnt = wsCnt[gid];
    const float denom = (float)(cnt > 1 ? cnt : 1);

    float* p = vox + (size_t)b * CC * R3 + vv;
    #pragma unroll 8
    for (int c = 0; c < CC; ++c) {
        const size_t off = (size_t)c * R3;
        p[off] = p[off] / denom;      // true division, matching sums / max(cnt,1)
    }
}

// ---------------------------------------------------------------------------
extern "C" void kernel_launch(void* const* d_in, const int* in_sizes, int n_in,
                              void* d_out, int out_size, void* d_ws, size_t ws_size,
                              hipStream_t stream) {
    const float* feats = (const float*)d_in[0];   // [16, 64, 65536]
    const float* pts   = (const float*)d_in[1];   // [16, 65536, 3]

    float* out       = (float*)d_out;
    float* vox       = out;                       // [16, 64, 32, 32, 32]
    float* coordsOut = out + (size_t)VOX_TOTAL;   // [16, 3, 65536]

    // Workspace layout: flat voxel idx per point (4 MB) | int counts (2 MB) | means (48 f)
    int*   wsIdx  = (int*)d_ws;                   // BB*NN
    int*   wsCnt  = wsIdx + (size_t)BB * NN;      // BB*R3
    float* wsMean = (float*)(wsCnt + (size_t)BB * R3);

    // 0) zero accumulators (d_out / d_ws are poisoned by the harness)
    vx_zero_kernel<<<VOX_TOTAL / 4 / 256, 256, 0, stream>>>((float4*)vox, (int4*)wsCnt);

    // 1) per-batch means via WMMA f32 column-sum reduction
    vx_mean_kernel<<<BB, 256, 0, stream>>>(pts, wsMean);

    // 2) coords output + flat indices + counts
    vx_coords_kernel<<<(BB * NN) / 256, 256, 0, stream>>>(pts, wsMean, coordsOut,
                                                          wsIdx, wsCnt);

    // 3) feature scatter-add (fp32 atomics, L2-resident)
    vx_scatter_kernel<<<(BB * NN) / 256, 256, 0, stream>>>(feats, wsIdx, vox);

    // 4) normalize by counts
    vx_norm_kernel<<<(BB * R3) / 256, 256, 0, stream>>>(wsCnt, vox);
}